// Adder2D_69337952026918
// MI455X (gfx1250) — compile-verified
//
#include <hip/hip_runtime.h>
#include <stdint.h>

#define N_    16
#define CI_   64
#define CO_   64
#define HW_   32
#define K_    3
#define CHUNK 4
#define NCHUNK (CI_ / CHUNK)      // 16
#define ROWS  34                   // padded rows per channel
#define RSTR  36                   // floats per padded row (16B-aligned stride)
#define CHF   (ROWS * RSTR)        // 1224 floats per channel plane
#define BUFF  (CHUNK * CHF)        // 4896 floats per LDS buffer

struct alignas(16) Smem {
    float guard[4];                // stays zero; backs the col=-1 read at buffer 0 edge
    float xbuf[2][BUFF];           // double-buffered padded input chunks
    float wlds[CI_ * 12];          // 64 ci x 3 kh x 4 (kw0,kw1,kw2,pad)
};

__device__ __forceinline__ void async_ld_b128(uint32_t ldsOff, uint32_t gOff, const float* sbase) {
    // GLOBAL_LOAD_ASYNC_TO_LDS_B128, GVS mode: mem = SADDR + VADDR(u32) + IOFFSET
    // VDST holds the LDS byte offset. Tracked with ASYNCcnt.
    asm volatile("global_load_async_to_lds_b128 %0, %1, %2 offset:0"
                 :: "v"(ldsOff), "v"(gOff), "s"(sbase)
                 : "memory");
}

__device__ __forceinline__ void issue_chunk(const float* xn, uint32_t bufOff, int ci0,
                                            uint32_t ldsThr, uint32_t thrOff) {
#pragma unroll
    for (int i = 0; i < CHUNK; ++i) {
        uint32_t g = (uint32_t)(ci0 + i) * (HW_ * HW_ * 4u) + thrOff;   // bytes within x[n]
        uint32_t l = bufOff + (uint32_t)i * (CHF * 4u) + ldsThr;        // LDS byte offset
        async_ld_b128(l, g, xn);
    }
}

__global__ __launch_bounds__(256) void adder2d_kernel(const float* __restrict__ x,
                                                      const float* __restrict__ w,
                                                      float* __restrict__ out) {
    __shared__ Smem sm;
    const int tid = threadIdx.x;
    const int co  = blockIdx.x;
    const int n   = blockIdx.y;

    // ---- one-time: zero guard + halos (whole LDS block) ----
    {
        float4* z = (float4*)&sm;
        const int nvec = (int)(sizeof(Smem) / 16);
        for (int i = tid; i < nvec; i += 256) z[i] = float4{0.f, 0.f, 0.f, 0.f};
    }
    __syncthreads();

    // ---- weights for this co into LDS (padded rows of 4 for b128 broadcast reads) ----
    {
        const float* wp = w + (size_t)co * (CI_ * K_ * K_);
        for (int i = tid; i < CI_ * K_ * K_; i += 256) {
            int ci = i / 9, t = i % 9;
            sm.wlds[ci * 12 + (t / 3) * 4 + (t % 3)] = wp[i];
        }
    }

    const float* xn = x + (size_t)n * (CI_ * HW_ * HW_);
    const int row  = tid >> 3;            // 0..31
    const int xseg = (tid & 7) << 2;      // 0,4,...,28

    // async-load addressing (each thread ships one 16B group per channel of the chunk)
    const uint32_t thrOff = (uint32_t)(row * HW_ + xseg) * 4u;            // global bytes in-plane
    const uint32_t ldsThr = (uint32_t)((row + 1) * RSTR + xseg) * 4u;     // interior at (r+1, c)
    const uint32_t buf0 = (uint32_t)(size_t)&sm.xbuf[0][0];
    const uint32_t buf1 = (uint32_t)(size_t)&sm.xbuf[1][0];

    issue_chunk(xn, buf0, 0, ldsThr, thrOff);

    float acc0 = 0.f, acc1 = 0.f, acc2 = 0.f, acc3 = 0.f;
    const float4* wv = (const float4*)sm.wlds;

#pragma unroll 1
    for (int k = 0; k < NCHUNK; ++k) {
        // Preload this chunk's weights into VGPRs BEFORE the async wait/barrier:
        // wlds is written only in the prologue (no barrier follows that fill before the k=0 reads), and these ds_load_b128s
        // overlap the s_wait_asynccnt latency instead of stalling the tap loop.
        float4 wreg[CHUNK][K_];
#pragma unroll
        for (int c = 0; c < CHUNK; ++c)
#pragma unroll
            for (int kh = 0; kh < K_; ++kh)
                wreg[c][kh] = wv[(k * CHUNK + c) * 3 + kh];

        if (k + 1 < NCHUNK) {
            issue_chunk(xn, ((k + 1) & 1) ? buf1 : buf0, (k + 1) * CHUNK, ldsThr, thrOff);
            asm volatile("s_wait_asynccnt 0x4" ::: "memory");  // chunk k done, k+1 in flight
        } else {
            asm volatile("s_wait_asynccnt 0x0" ::: "memory");
        }
        __syncthreads();  // chunk k visible to all waves

        const float* buf = &sm.xbuf[k & 1][0];
#pragma unroll
        for (int c = 0; c < CHUNK; ++c) {
            const float* chb = buf + c * CHF;

            // Phase 1: batch all LDS reads for this channel (3 halo rows x 6 values)
            float v[K_][6];
#pragma unroll
            for (int kh = 0; kh < K_; ++kh) {
                const float* p = chb + (row + kh) * RSTR + xseg;
                const float  l = p[-1];                 // halo/guard (zero at edges)
                const float4 m = *(const float4*)p;     // aligned ds_load_b128
                const float  r = p[4];
                v[kh][0] = l;   v[kh][1] = m.x; v[kh][2] = m.y;
                v[kh][3] = m.z; v[kh][4] = m.w; v[kh][5] = r;
            }

            // Phase 2: pure VALU — 9 taps x 4 outputs, 2 ops/tap
#pragma unroll
            for (int kh = 0; kh < K_; ++kh) {
                const float4 wk = wreg[c][kh];
                acc0 -= fabsf(v[kh][0] - wk.x) + fabsf(v[kh][1] - wk.y) + fabsf(v[kh][2] - wk.z);
                acc1 -= fabsf(v[kh][1] - wk.x) + fabsf(v[kh][2] - wk.y) + fabsf(v[kh][3] - wk.z);
                acc2 -= fabsf(v[kh][2] - wk.x) + fabsf(v[kh][3] - wk.y) + fabsf(v[kh][4] - wk.z);
                acc3 -= fabsf(v[kh][3] - wk.x) + fabsf(v[kh][4] - wk.y) + fabsf(v[kh][5] - wk.z);
            }
        }
        __syncthreads();  // everyone done reading buf[k&1] before it is refilled
    }

    float4 o = {acc0, acc1, acc2, acc3};
    *(float4*)(out + (((size_t)(n * CO_ + co) * HW_ + row) * HW_ + xseg)) = o;
}

extern "C" void kernel_launch(void* const* d_in, const int* in_sizes, int n_in,
                              void* d_out, int out_size, void* d_ws, size_t ws_size,
                              hipStream_t stream) {
    (void)in_sizes; (void)n_in; (void)out_size; (void)d_ws; (void)ws_size;
    const float* x = (const float*)d_in[0];
    const float* w = (const float*)d_in[1];
    float* out = (float*)d_out;
    dim3 grid(CO_, N_);
    adder2d_kernel<<<grid, dim3(256), 0, stream>>>(x, w, out);
}